// LennardJonesCluster_6313601925568
// MI455X (gfx1250) — compile-verified
//
#include <hip/hip_runtime.h>

typedef __attribute__((ext_vector_type(2))) float v2f;
typedef __attribute__((ext_vector_type(8))) float v8f;

#define G_NUM 256
#define A_NUM 192
#define CUT2 6.25f
#define NWAVES 12
#define BLOCK (NWAVES * 32)

// One workgroup per graph. Wave w owns 16-row block I=w of the 192x192 pair
// matrix and sweeps the 12 column blocks. Each 16x16 tile's squared-distance
// matrix comes from one V_WMMA_F32_16X16X4_F32 (Gram trick); the LJ r^-6
// chemistry runs on VALU and co-executes with the matrix pipe.
__global__ __launch_bounds__(BLOCK) void lj_wmma_kernel(const float* __restrict__ pos,
                                                        float* __restrict__ energy_out,
                                                        float* __restrict__ force_out) {
  __shared__ float xs[A_NUM], ys[A_NUM], zs[A_NUM], qs[A_NUM];
  __shared__ float ered[NWAVES];

  const int g    = blockIdx.x;
  const int tid  = threadIdx.x;
  const int lane = tid & 31;
  const int m    = lane & 15;   // column within tile (and A/B row within half)
  const int half = lane >> 4;   // 0: lanes 0-15, 1: lanes 16-31

  // Wave index pinned to an SGPR so tile-index compares and base addressing
  // are scalar (s_cmp/s_cbranch), not per-lane v_cmp + cndmask.
  const int I = __builtin_amdgcn_readfirstlane(tid >> 5);   // row block 0..11

  // Stage this graph's atoms into LDS (2.3 KB — whole graph is LDS-resident).
  if (tid < A_NUM) {
    const float* p = pos + (size_t)(g * A_NUM + tid) * 3;
    float x = p[0], y = p[1], z = p[2];
    xs[tid] = x; ys[tid] = y; zs[tid] = z;
    qs[tid] = x * x + y * y + z * z;   // |p|^2 for the Gram trick
  }
  __syncthreads();

  const int ibase = I * 16;

  // A fragment (f32 16x4, ISA layout): lanes 0-15 carry (K0,K1)=(x,y),
  // lanes 16-31 carry (K2,K3)=(z,q) for row M = lane&15.
  v2f afrag;
  if (half == 0) { afrag.x = xs[ibase + m]; afrag.y = ys[ibase + m]; }
  else           { afrag.x = zs[ibase + m]; afrag.y = qs[ibase + m]; }

  // Row-position constants for the 8 rows this lane accumulates (M = r + 8*half),
  // and the diagonal "bump": on the J==I tile, self pairs get d2 += 1e30 so the
  // cutoff test kills them — no per-element index compares in the hot loop.
  float xi[8], yi[8], zi[8], bump[8];
#pragma unroll
  for (int r = 0; r < 8; ++r) {
    int row = ibase + r + 8 * half;
    xi[r] = xs[row]; yi[r] = ys[row]; zi[r] = zs[row];
    bump[r] = ((r + 8 * half) == m) ? 1e30f : 0.0f;
  }

  float fx[8], fy[8], fz[8];
#pragma unroll
  for (int r = 0; r < 8; ++r) { fx[r] = 0.f; fy[r] = 0.f; fz[r] = 0.f; }
  float eacc = 0.f;

#pragma unroll 4
  for (int J = 0; J < 12; ++J) {
    const int jb = J * 16;
    const float xj = xs[jb + m], yj = ys[jb + m], zj = zs[jb + m];
    const float qj = qs[jb + m];
    // B fragment (f32 4x16): column N = lane&15 holds [-2x, -2y, -2z, 1].
    v2f bfrag;
    if (half == 0) { bfrag.x = -2.0f * xj; bfrag.y = -2.0f * yj; }
    else           { bfrag.x = -2.0f * zj; bfrag.y = 1.0f; }

    v8f cz = {};
    // D_ij = -2 p_i . p_j + q_i   (one matrix op per 16x16 tile)
    v8f d2t = __builtin_amdgcn_wmma_f32_16x16x4_f32(
        /*neg_a=*/false, afrag, /*neg_b=*/false, bfrag,
        /*c_mod=*/(short)0, cz, /*reuse_a=*/false, /*reuse_b=*/false);

    if (J == I) {              // scalar branch: one tile in 12 pays the mask
#pragma unroll
      for (int r = 0; r < 8; ++r) d2t[r] += bump[r];
    }

#pragma unroll
    for (int r = 0; r < 8; ++r) {
      float d2 = d2t[r] + qj;                 // r^2 = q_i + q_j - 2 p_i.p_j
      const bool valid = (d2 < CUT2);         // self pairs bumped out of range
      d2 = fmaxf(d2, 1e-6f);                  // guard Gram cancellation at tiny r
      // Fast reciprocal: v_rcp_f32 + one Newton step (~1 ulp, vs 10-op IEEE div)
      float inv2 = __builtin_amdgcn_rcpf(d2);
      inv2 = fmaf(fmaf(-d2, inv2, 1.0f), inv2, inv2);
      const float inv6  = inv2 * inv2 * inv2;
      const float inv12 = inv6 * inv6;
      const float e = valid ? 4.0f * (inv12 - inv6) : 0.0f;
      const float s = valid ? 24.0f * (2.0f * inv12 - inv6) * inv2 : 0.0f;
      eacc += e;
      fx[r] = fmaf(s, xi[r] - xj, fx[r]);
      fy[r] = fmaf(s, yi[r] - yj, fy[r]);
      fz[r] = fmaf(s, zi[r] - zj, fz[r]);
    }
  }

  // Deterministic energy reduction: wave -> LDS -> thread 0 (no float atomics).
#pragma unroll
  for (int off = 16; off >= 1; off >>= 1) eacc += __shfl_xor(eacc, off, 32);
  if (lane == 0) ered[I] = eacc;
  __syncthreads();
  if (tid == 0) {
    float t = 0.f;
    for (int w = 0; w < NWAVES; ++w) t += ered[w];
    energy_out[g] = 0.5f * t;   // full matrix counts each pair twice
  }

  // Force reduction across each 16-lane half (xor masks 8..1 stay in-half on wave32).
#pragma unroll
  for (int off = 8; off >= 1; off >>= 1) {
#pragma unroll
    for (int r = 0; r < 8; ++r) {
      fx[r] += __shfl_xor(fx[r], off, 32);
      fy[r] += __shfl_xor(fy[r], off, 32);
      fz[r] += __shfl_xor(fz[r], off, 32);
    }
  }
  if (m == 0) {                 // lanes 0 and 16 own rows r and 8+r
#pragma unroll
    for (int r = 0; r < 8; ++r) {
      const int irow = ibase + r + 8 * half;
      const size_t a = (size_t)(g * A_NUM + irow) * 3;
      force_out[a + 0] = fx[r];
      force_out[a + 1] = fy[r];
      force_out[a + 2] = fz[r];
    }
  }
}

extern "C" void kernel_launch(void* const* d_in, const int* in_sizes, int n_in,
                              void* d_out, int out_size, void* d_ws, size_t ws_size,
                              hipStream_t stream) {
  const float* pos = (const float*)d_in[0];   // [G*A, 3] f32
  // d_in[1] (batch) is constant metadata: contiguous equal blocks, unused.
  float* out = (float*)d_out;                 // [G] energy ++ [G*A,3] forces
  lj_wmma_kernel<<<G_NUM, BLOCK, 0, stream>>>(pos, out, out + G_NUM);
}